// CRFSmooth2D_24189255811166
// MI455X (gfx1250) — compile-verified
//
#include <hip/hip_runtime.h>
#include <hip/hip_bf16.h>

#define B_ 8
#define K_ 16
#define C_ 3
#define H_ 512
#define W_ 512
#define HW_ (H_ * W_)

typedef float v2f __attribute__((ext_vector_type(2)));
typedef float v8f __attribute__((ext_vector_type(8)));

// ---- CDNA5 async global->LDS DMA (ASYNCcnt-tracked, no VGPR round trip) ----
__device__ __forceinline__ unsigned lds_off(const void* p) {
  // LDS aperture: flat addr[31:0] == byte offset from LDS_BASE
  return (unsigned)(size_t)p;
}
__device__ __forceinline__ void async_load_b128(const float* g, unsigned loff) {
  asm volatile("global_load_async_to_lds_b128 %0, %1, off"
               :
               : "v"(loff), "v"(g)
               : "memory");
}
__device__ __forceinline__ void async_load_b32(const float* g, unsigned loff) {
  asm volatile("global_load_async_to_lds_b32 %0, %1, off"
               :
               : "v"(loff), "v"(g)
               : "memory");
}
__device__ __forceinline__ void wait_async0() {
  asm volatile("s_wait_asynccnt 0x0" ::: "memory");
}

// ---------------- Kernel A: per-(b,k) reductions via V_WMMA_F32_16X16X4_F32 ---
// D(16x16) accumulates rows {x0,x1,x2,1} x 16 classes, K-dim = pixels.
#define A_BLOCKS_PER_B 64
#define A_PPB (HW_ / A_BLOCKS_PER_B)  // 4096 pixels per block
#define A_CHUNK 512                   // pixels staged per iteration
#define LPITCH 532                    // label row pitch (16B aligned, conflict-free)

__global__ __launch_bounds__(256) void crf_reduce_kernel(
    const float* __restrict__ labels, const float* __restrict__ inputs,
    float* __restrict__ partials) {
  __shared__ float lab_s[K_ * LPITCH];   // [class][pixel]
  __shared__ float inp_s[3 * A_CHUNK];   // [channel][pixel]
  __shared__ float red_s[8 * 64];

  const int b    = blockIdx.x >> 6;
  const int blk  = blockIdx.x & 63;
  const int tid  = threadIdx.x;
  const int lane = tid & 31;
  const int wv   = tid >> 5;
  const int base = blk * A_PPB;

  const float* inb = inputs + (size_t)b * C_ * HW_;
  const float* lb  = labels + (size_t)b * K_ * HW_;

  const int m  = lane & 15;  // A row (channels 0..2, ones row 3) == B class lane
  const int hi = lane >> 4;  // low/high half of wave

  v8f acc = {};  // rows 0..3 x 16 classes accumulate here for all pixels

  for (int it = 0; it < A_PPB / A_CHUNK; ++it) {
    const int cbase = base + it * A_CHUNK;
    // async DMA: 16 label rows, 128b per lane, straight into LDS
    for (int q = tid; q < K_ * (A_CHUNK / 4); q += 256) {
      const int k  = q >> 7;            // 128 float4 per row
      const int p4 = (q & 127) << 2;
      async_load_b128(lb + (size_t)k * HW_ + cbase + p4,
                      lds_off(&lab_s[k * LPITCH + p4]));
    }
    // async DMA: 3 input channel rows
    for (int q = tid; q < 3 * (A_CHUNK / 4); q += 256) {
      const int c  = q >> 7;
      const int p4 = (q & 127) << 2;
      async_load_b128(inb + (size_t)c * HW_ + cbase + p4,
                      lds_off(&inp_s[c * A_CHUNK + p4]));
    }
    wait_async0();
    __syncthreads();

    const int wbase = wv * 64;  // 64 pixels per wave per chunk -> 16 WMMAs
    for (int g = 0; g < 16; ++g) {
      const int p0 = wbase + g * 4;
      const int pa = p0 + (hi << 1);  // A K-cols: lanes<16 -> {0,1}, >=16 -> {2,3}
      v2f av;  // A[m][kk]: channel m at pixels pa, pa+1; row3 = ones
      if (m < 3) {
        av.x = inp_s[m * A_CHUNK + pa];
        av.y = inp_s[m * A_CHUNK + pa + 1];
      } else if (m == 3) {
        av.x = 1.0f;
        av.y = 1.0f;
      } else {
        av.x = 0.0f;
        av.y = 0.0f;
      }
      v2f bv;  // B[kk][class]: labels of class (lane&15) at pixels pa, pa+1
      bv.x = lab_s[m * LPITCH + pa];
      bv.y = lab_s[m * LPITCH + pa + 1];
      acc = __builtin_amdgcn_wmma_f32_16x16x4_f32(false, av, false, bv,
                                                  (short)0, acc, false, false);
    }
    __syncthreads();
  }

  // cross-wave reduce: D layout -> VGPR v holds row v (lanes 0..15)
  if (lane < 16) {
    red_s[wv * 64 + 0 * 16 + lane] = acc[0];
    red_s[wv * 64 + 1 * 16 + lane] = acc[1];
    red_s[wv * 64 + 2 * 16 + lane] = acc[2];
    red_s[wv * 64 + 3 * 16 + lane] = acc[3];
  }
  __syncthreads();
  if (tid < 64) {
    float s = 0.0f;
    for (int w = 0; w < 8; ++w) s += red_s[w * 64 + tid];
    partials[(size_t)blockIdx.x * 64 + tid] = s;  // slot [m*16 + k]
  }
}

// ---------------- Kernel B: fold partials -> class means + m2 ----------------
__global__ void crf_finalize_kernel(const float* __restrict__ partials,
                                    float* __restrict__ fin) {
  const int t = threadIdx.x;
  if (t >= B_ * K_) return;
  const int b = t >> 4, k = t & 15;
  float d0 = 0.f, d1 = 0.f, d2 = 0.f, d3 = 0.f;
  for (int blk = 0; blk < A_BLOCKS_PER_B; ++blk) {
    const float* p = partials + ((size_t)(b * A_BLOCKS_PER_B + blk)) * 64;
    d0 += p[0 * 16 + k];
    d1 += p[1 * 16 + k];
    d2 += p[2 * 16 + k];
    d3 += p[3 * 16 + k];
  }
  const float inv = 1.0f / (float)HW_;
  const float den = d3 * inv + 1e-5f;
  const float mu0 = d0 * inv / den;
  const float mu1 = d1 * inv / den;
  const float mu2 = d2 * inv / den;
  fin[t * 4 + 0] = mu0;
  fin[t * 4 + 1] = mu1;
  fin[t * 4 + 2] = mu2;
  fin[t * 4 + 3] = mu0 * mu0 + mu1 * mu1 + mu2 * mu2;
}

// ---------------- Kernel C: fused w-field (WMMA) + 3x3 depthwise conv --------
#define TS 32
#define HS 34
#define NP (HS * HS)  // 1156 halo pixels

__global__ __launch_bounds__(256) void crf_smooth_kernel(
    const float* __restrict__ labels, const float* __restrict__ inputs,
    const float* __restrict__ fin, float* __restrict__ out) {
  __shared__ float inp_s[3 * NP];   // [channel][pixel]
  __shared__ float lab_s[K_ * NP];  // [class][pixel]
  __shared__ float w_s[NP * 17];    // [pixel][class], pad 17

  const int b    = blockIdx.z;
  const int ty0  = blockIdx.y * TS;
  const int tx0  = blockIdx.x * TS;
  const int tid  = threadIdx.x;
  const int lane = tid & 31;
  const int wv   = tid >> 5;

  const float* inb = inputs + (size_t)b * C_ * HW_;
  const float* lb  = labels + (size_t)b * K_ * HW_;

  // stage tile+halo: EXEC-masked async DMA for interior, zero-fill outside image
  for (int idx = tid; idx < NP; idx += 256) {
    const int py = idx / HS, px = idx - py * HS;
    const int gh = ty0 + py - 1, gw = tx0 + px - 1;
    const bool in = (unsigned)gh < H_ && (unsigned)gw < W_;
    const size_t o = (size_t)gh * W_ + gw;
    for (int c = 0; c < 3; ++c) {
      if (in)
        async_load_b32(inb + (size_t)c * HW_ + o, lds_off(&inp_s[c * NP + idx]));
      else
        inp_s[c * NP + idx] = 0.f;
    }
    for (int k = 0; k < K_; ++k) {
      if (in)
        async_load_b32(lb + (size_t)k * HW_ + o, lds_off(&lab_s[k * NP + idx]));
      else
        lab_s[k * NP + idx] = 0.f;
    }
  }

  // per-lane class constants: lane = class in B/D layouts
  const int kc = lane & 15;
  const int hi = lane >> 4;
  const float* fb = fin + (size_t)(b * K_ + kc) * 4;
  const float m2 = fb[3];
  v2f bv;  // B[kk][class] = {-2mu0,-2mu1,-2mu2,1}
  bv.x = -2.0f * fb[hi ? 2 : 0];
  bv.y = hi ? 1.0f : -2.0f * fb[1];

  wait_async0();
  __syncthreads();

  // w = exp(-4*diff^2); diff via one WMMA per 16-pixel group
  const int m = lane & 15;
  for (int g = wv; g * 16 < NP; g += 8) {
    const int gb = g * 16;
    int pm = gb + m;
    if (pm >= NP) pm = NP - 1;  // clamp (stores below are masked)
    const float x0 = inp_s[0 * NP + pm];
    const float x1 = inp_s[1 * NP + pm];
    const float x2 = inp_s[2 * NP + pm];
    v2f av;  // A[pixel m][kk]: {x0,x1} lanes<16, {x2, x2sum} lanes>=16
    av.x = hi ? x2 : x0;
    av.y = hi ? (x0 * x0 + x1 * x1 + x2 * x2) : x1;
    v8f acc = {};
    acc = __builtin_amdgcn_wmma_f32_16x16x4_f32(false, av, false, bv,
                                                (short)0, acc, false, false);
#pragma unroll
    for (int v = 0; v < 8; ++v) {
      const int pix = gb + v + hi * 8;  // D: VGPR v = rows v / v+8
      if (pix < NP) {
        const float diff = acc[v] + m2;
        const int py = pix / HS, px = pix - py * HS;
        const int gh = ty0 + py - 1, gw = tx0 + px - 1;
        float wt = 0.0f;  // zero weight outside image == 'same' zero padding
        if ((unsigned)gh < H_ && (unsigned)gw < W_)
          wt = __expf(-4.0f * diff * diff);
        w_s[pix * 17 + kc] = wt;
      }
    }
  }
  __syncthreads();

  // separable Gaussian taps: e = {exp(-2), 1, exp(-2)}
  const float KW0 = 0.13533528323661270f;
  const float e3[3] = {KW0, 1.0f, KW0};

  for (int k = 0; k < K_; ++k) {
    const float* lks = lab_s + k * NP;
    for (int t = tid; t < TS * TS; t += 256) {
      const int ty = t >> 5, tx = t & 31;
      const int hp = (ty + 1) * HS + (tx + 1);
      float num = 0.f, den = 0.f;
#pragma unroll
      for (int dy = -1; dy <= 1; ++dy) {
#pragma unroll
        for (int dx = -1; dx <= 1; ++dx) {
          const int q = hp + dy * HS + dx;
          const float kv = e3[dy + 1] * e3[dx + 1];
          const float wq = w_s[q * 17 + k];
          den += kv * wq;
          num += kv * wq * lks[q];
        }
      }
      out[(((size_t)b * K_ + k) * H_ + (ty0 + ty)) * W_ + (tx0 + tx)] =
          num / (den + 1e-6f);
    }
  }
}

extern "C" void kernel_launch(void* const* d_in, const int* in_sizes, int n_in,
                              void* d_out, int out_size, void* d_ws, size_t ws_size,
                              hipStream_t stream) {
  const float* labels = (const float*)d_in[0];  // [B,K,H,W]
  const float* inputs = (const float*)d_in[1];  // [B,C,H,W]
  float* out = (float*)d_out;                   // [B,K,H,W]

  float* partials = (float*)d_ws;                            // B*64*64 floats
  float* fin = partials + (size_t)B_ * A_BLOCKS_PER_B * 64;  // B*K*4 floats

  crf_reduce_kernel<<<dim3(B_ * A_BLOCKS_PER_B), 256, 0, stream>>>(labels, inputs,
                                                                   partials);
  crf_finalize_kernel<<<dim3(1), 128, 0, stream>>>(partials, fin);
  crf_smooth_kernel<<<dim3(W_ / TS, H_ / TS, B_), 256, 0, stream>>>(labels, inputs,
                                                                    fin, out);
}